// Complex_CA_18356690224015
// MI455X (gfx1250) — compile-verified
//
#include <hip/hip_runtime.h>
#include <hip/hip_bf16.h>

#define GDIM   34
#define NPIX   (GDIM*GDIM)      // 1156
#define PSTR   36               // padded stride (1-ring of zeros)
#define PSZ    (PSTR*PSTR)      // 1296
#define NTILES 73               // ceil(1156/16)
#define BLOCK  256              // 8 wave32 waves

typedef float v2f __attribute__((ext_vector_type(2)));
typedef float v8f __attribute__((ext_vector_type(8)));

__device__ __forceinline__ v8f wmma4(v2f a, v2f b, v8f c) {
  // D = A(16x4 f32) * B(4x16 f32) + C(16x16 f32)
  return __builtin_amdgcn_wmma_f32_16x16x4_f32(false, a, false, b, (short)0, c,
                                               false, false);
}

// flat pixel p (row*34+col) -> index in zero-padded 36x36 layout
__device__ __forceinline__ int pidx(unsigned p) {
  return (int)(p + 2u * (p / 34u) + 37u);
}

__device__ __forceinline__ float max9(const float* b, int i) {
  float m = fmaxf(b[i - 37], b[i - 36]);
  m = fmaxf(m, b[i - 35]); m = fmaxf(m, b[i - 1]); m = fmaxf(m, b[i]);
  m = fmaxf(m, b[i + 1]);  m = fmaxf(m, b[i + 35]); m = fmaxf(m, b[i + 36]);
  return fmaxf(m, b[i + 37]);
}

__global__ __launch_bounds__(BLOCK) void Complex_CA_kernel(
    const float* __restrict__ cell_in, const float* __restrict__ food_in,
    const float* __restrict__ fc1w, const float* __restrict__ fc1b,
    const float* __restrict__ fc2w, const float* __restrict__ fc2b,
    const float* __restrict__ skern, const int* __restrict__ steps_p,
    float* __restrict__ out, int Bn) {
  __shared__ float sbufA[4 * PSZ];       // state buffer A (padded, zero border)
  __shared__ float sbufB[4 * PSZ];       // state buffer B
  __shared__ float sscent[NPIX];         // precomputed scent field
  __shared__ float sh_h[8][16 * 64];     // per-wave fc1 activations (D->A relayout)
  __shared__ float smask[NPIX];          // combined pre&post mask (also food temp)
  __shared__ unsigned shist[256];        // radix-select histogram
  __shared__ float skl[19 * 19];         // 19x19 scent kernel
  __shared__ unsigned s_cl, s_rank, s_prefix, s_liv;
  __shared__ float s_tot;

  const int tid  = threadIdx.x;
  const int b    = blockIdx.x;
  const int lane = tid & 31;
  const int wave = tid >> 5;
  const int half = lane >> 4;   // 0: lanes 0-15, 1: lanes 16-31
  const int nlo  = lane & 15;

  // ---------------- init ----------------
  for (int i = tid; i < 4 * PSZ; i += BLOCK) { sbufA[i] = 0.f; sbufB[i] = 0.f; }
  for (int i = tid; i < 361; i += BLOCK) skl[i] = skern[i];
  for (unsigned p = tid; p < NPIX; p += BLOCK)
    smask[p] = food_in[(size_t)b * NPIX + p];   // stash food in mask buffer
  if (tid == 0) { s_cl = 0u; s_liv = 0u; s_tot = 0.f; }
  __syncthreads();

  // load cell state into padded interior of bufA
  for (unsigned p = tid; p < NPIX; p += BLOCK) {
    int i = pidx(p);
#pragma unroll
    for (int c = 0; c < 4; ++c)
      sbufA[c * PSZ + i] = cell_in[((size_t)b * 4 + c) * NPIX + p];
  }
  // one-time 19x19 scent convolution (zero padding 9)
  for (unsigned p = tid; p < NPIX; p += BLOCK) {
    int row = (int)(p / 34u), col = (int)(p % 34u);
    float acc = 0.f;
    for (int u = 0; u < 19; ++u) {
      int r2 = row + u - 9;
      if (r2 < 0 || r2 >= GDIM) continue;
      for (int v = 0; v < 19; ++v) {
        int c2 = col + v - 9;
        if (c2 < 0 || c2 >= GDIM) continue;
        acc += smask[r2 * GDIM + c2] * skl[u * 19 + v];
      }
    }
    sscent[p] = acc;
  }

  // ---- loop-invariant WMMA B fragments (weights) in registers ----
  // B layout (4x16): lanes0-15 hold rows K=4kb+0 (.x), 4kb+1 (.y) at N=lane;
  //                  lanes16-31 hold rows K=4kb+2 (.x), 4kb+3 (.y) at N=lane-16.
  v2f B1[3][4];
  float bias1[4];
#pragma unroll
  for (int kb = 0; kb < 3; ++kb)
#pragma unroll
    for (int nt = 0; nt < 4; ++nt) {
      int k = 4 * kb + 2 * half;
      int n = nt * 16 + nlo;
      B1[kb][nt].x = fc1w[n * 12 + k];
      B1[kb][nt].y = fc1w[n * 12 + k + 1];
    }
#pragma unroll
  for (int nt = 0; nt < 4; ++nt) bias1[nt] = fc1b[nt * 16 + nlo];

  v2f B2[16];
  float bias2 = (nlo < 4) ? fc2b[nlo] : 0.f;
#pragma unroll
  for (int kb = 0; kb < 16; ++kb) {
    int k = 4 * kb + 2 * half;
    B2[kb].x = (nlo < 4) ? fc2w[nlo * 64 + k] : 0.f;
    B2[kb].y = (nlo < 4) ? fc2w[nlo * 64 + k + 1] : 0.f;
  }

  float* cur = sbufA;
  float* nxt = sbufB;
  const int steps = steps_p[0];
  __syncthreads();

  // ---------------- step loop ----------------
  for (int s = 0; s < steps; ++s) {
    // Phase A: cell[3] = scent, count living (ch0 > 0.8)
    unsigned cnt = 0;
    for (unsigned p = tid; p < NPIX; p += BLOCK) {
      int i = pidx(p);
      cur[3 * PSZ + i] = sscent[p];
      if (cur[i] > 0.8f) cnt++;
    }
    atomicAdd(&s_cl, cnt);
    __syncthreads();
    const unsigned cl = s_cl;

    // Phase B: perception + MLP via WMMA, write x = cell + upd into nxt
    for (int t = wave; t < NTILES; t += 8) {
      unsigned pix = (unsigned)t * 16 + nlo;
      bool valid = pix < NPIX;
      int ci = pidx(valid ? pix : 0u);
      const float* p0 = cur + (2 * half) * PSZ;  // channel 2*half
      const float* p1 = p0 + PSZ;                // channel 2*half+1
      float a00 = p0[ci - 37], a01 = p0[ci - 36], a02 = p0[ci - 35];
      float a10 = p0[ci - 1],  a11 = p0[ci],      a12 = p0[ci + 1];
      float a20 = p0[ci + 35], a21 = p0[ci + 36], a22 = p0[ci + 37];
      float c00 = p1[ci - 37], c01 = p1[ci - 36], c02 = p1[ci - 35];
      float c10 = p1[ci - 1],  c11 = p1[ci],      c12 = p1[ci + 1];
      float c20 = p1[ci + 35], c21 = p1[ci + 36], c22 = p1[ci + 37];
      float sx0 = (a02 - a00 + 2.f * (a12 - a10) + a22 - a20) * 0.125f;
      float sy0 = (a20 - a00 + 2.f * (a21 - a01) + a22 - a02) * 0.125f;
      float sx1 = (c02 - c00 + 2.f * (c12 - c10) + c22 - c20) * 0.125f;
      float sy1 = (c20 - c00 + 2.f * (c21 - c01) + c22 - c02) * 0.125f;
      v2f A0, A1, A2;  // A layout: lane m holds M=m, .x/.y = K pair per half
      A0.x = valid ? a11 : 0.f; A0.y = valid ? c11 : 0.f;  // raw channels
      A1.x = valid ? sx0 : 0.f; A1.y = valid ? sx1 : 0.f;  // sobel-x
      A2.x = valid ? sy0 : 0.f; A2.y = valid ? sy1 : 0.f;  // sobel-y
      float* hp = sh_h[wave];
      // fc1: [16x12] x [12x64] -> relu -> h, 4 N-tiles, 3 K-steps each
#pragma unroll
      for (int nt = 0; nt < 4; ++nt) {
        v8f acc;
#pragma unroll
        for (int r = 0; r < 8; ++r) acc[r] = bias1[nt];
        acc = wmma4(A0, B1[0][nt], acc);
        acc = wmma4(A1, B1[1][nt], acc);
        acc = wmma4(A2, B1[2][nt], acc);
        // D layout -> LDS: VGPR r holds M = r + 8*half, N = nlo
#pragma unroll
        for (int r = 0; r < 8; ++r)
          hp[(r + 8 * half) * 64 + nt * 16 + nlo] = fmaxf(acc[r], 0.f);
      }
      // fc2: [16x64] x [64x4pad16], 16 chained K-steps; A refetched from LDS
      v8f acc2;
#pragma unroll
      for (int r = 0; r < 8; ++r) acc2[r] = bias2;
#pragma unroll
      for (int kb = 0; kb < 16; ++kb) {
        v2f Af;
        Af.x = hp[nlo * 64 + 4 * kb + 2 * half];
        Af.y = hp[nlo * 64 + 4 * kb + 2 * half + 1];
        acc2 = wmma4(Af, B2[kb], acc2);
      }
      // scatter upd: lane with N=nlo<4 owns channel nlo of pixels r+8*half
      if (nlo < 4) {
#pragma unroll
        for (int r = 0; r < 8; ++r) {
          unsigned pm = (unsigned)t * 16 + r + 8 * half;
          if (pm < NPIX) {
            int ii = pidx(pm);
            nxt[nlo * PSZ + ii] = cur[nlo * PSZ + ii] + acc2[r];
          }
        }
      }
    }
    __syncthreads();

    // Phase C: pre/post 3x3 maxpool masks (zero-pad safe for >0.1 threshold)
    for (unsigned p = tid; p < NPIX; p += BLOCK) {
      int i = pidx(p);
      bool pre  = max9(cur, i) > 0.1f;
      bool post = max9(nxt, i) > 0.1f;
      smask[p] = (pre && post) ? 1.f : 0.f;
    }
    __syncthreads();

    // Phase D: apply mask + clips in place on nxt; reset selection state
    for (unsigned p = tid; p < NPIX; p += BLOCK) {
      int i = pidx(p);
      float mk = smask[p];
#pragma unroll
      for (int c = 0; c < 4; ++c) {
        float v = nxt[c * PSZ + i] * mk;
        v = fminf(fmaxf(v, -10.f), 10.f);
        if (c == 0) v = fminf(fmaxf(v, 0.f), 1.f);
        nxt[c * PSZ + i] = v;
      }
    }
    if (tid == 0) {
      s_cl = 0u;
      s_rank = (cl < NPIX) ? cl : (NPIX - 1);  // take_along_axis clamps OOB
      s_prefix = 0u;
    }
    __syncthreads();

    // Phase E: exact radix-select of rank-th smallest of nxt ch0 (all >= 0)
    const unsigned hm[4] = {0u, 0xFF000000u, 0xFFFF0000u, 0xFFFFFF00u};
#pragma unroll 1
    for (int pass = 0; pass < 4; ++pass) {
      int shift = 24 - 8 * pass;
      shist[tid] = 0u;
      __syncthreads();
      unsigned pref = s_prefix;
      for (unsigned p = tid; p < NPIX; p += BLOCK) {
        unsigned key = __float_as_uint(nxt[pidx(p)]);
        if (((key ^ pref) & hm[pass]) == 0u)
          atomicAdd(&shist[(key >> shift) & 255u], 1u);
      }
      __syncthreads();
      if (tid < 32) {  // wave0: 256-bin scan via 8 bins/lane + shfl scan
        unsigned v[8], ssum = 0u;
#pragma unroll
        for (int i = 0; i < 8; ++i) { v[i] = shist[tid * 8 + i]; ssum += v[i]; }
        unsigned incl = ssum;
#pragma unroll
        for (int off = 1; off < 32; off <<= 1) {
          unsigned u = __shfl_up(incl, off, 32);
          if (lane >= off) incl += u;
        }
        unsigned excl = incl - ssum;
        unsigned rk = s_rank;
        if (rk >= excl && rk < incl) {
          unsigned cum = excl;
          int bi = -1;
#pragma unroll
          for (int i = 0; i < 8; ++i) {
            if (bi < 0) {
              if (rk < cum + v[i]) bi = i; else cum += v[i];
            }
          }
          s_prefix = pref | ((unsigned)(tid * 8 + bi) << shift);
          s_rank = rk - cum;
        }
      }
      __syncthreads();
    }
    const float kth = __uint_as_float(s_prefix);
    // keep strictly greater than kth
    for (unsigned p = tid; p < NPIX; p += BLOCK) {
      int i = pidx(p);
      float v = nxt[i];
      nxt[i] = (v > kth) ? v : 0.f;
    }
    // swap ping-pong buffers
    float* tmp = cur; cur = nxt; nxt = tmp;
    __syncthreads();
  }

  // ---------------- epilogue: outputs ----------------
  const size_t OFF_FOOD = (size_t)Bn * 4 * NPIX;
  const size_t OFF_TOT  = OFF_FOOD + (size_t)Bn * NPIX;
  const size_t OFF_LIV  = OFF_TOT + (size_t)Bn;
  float lt = 0.f;
  unsigned lc = 0u;
  for (unsigned p = tid; p < NPIX; p += BLOCK) {
    int i = pidx(p);
#pragma unroll
    for (int c = 0; c < 4; ++c)
      out[((size_t)b * 4 + c) * NPIX + p] = cur[c * PSZ + i];
    out[OFF_FOOD + (size_t)b * NPIX + p] = food_in[(size_t)b * NPIX + p];
    float v = cur[i];
    lt += v;
    if (v > 0.1f) lc++;
  }
  atomicAdd(&s_tot, lt);
  atomicAdd(&s_liv, lc);
  __syncthreads();
  if (tid == 0) {
    out[OFF_TOT + b] = s_tot;
    out[OFF_LIV + b] = (float)lc * 0.f + (float)s_liv;
  }
}

extern "C" void kernel_launch(void* const* d_in, const int* in_sizes, int n_in,
                              void* d_out, int out_size, void* d_ws, size_t ws_size,
                              hipStream_t stream) {
  const float* cell  = (const float*)d_in[0];
  const float* food  = (const float*)d_in[1];
  const float* fc1w  = (const float*)d_in[2];
  const float* fc1b  = (const float*)d_in[3];
  const float* fc2w  = (const float*)d_in[4];
  const float* fc2b  = (const float*)d_in[5];
  const float* sk    = (const float*)d_in[6];
  const int*   steps = (const int*)d_in[7];
  float* out = (float*)d_out;
  int Bn = in_sizes[0] / (4 * NPIX);  // 1024
  hipLaunchKernelGGL(Complex_CA_kernel, dim3(Bn), dim3(BLOCK), 0, stream,
                     cell, food, fc1w, fc1b, fc2w, fc2b, sk, steps, out, Bn);
}